// SeperableResidualBlock_70884140253236
// MI455X (gfx1250) — compile-verified
//
#include <hip/hip_runtime.h>
#include <stdint.h>

// ---------------------------------------------------------------------------
// SeparableResidualBlock for MI455X (gfx1250), fp32 WMMA (16x16x4) everywhere.
// N=200000 voxels, A=128 in-ch, BC=256 out-ch, K=27 offsets, 8 branches of 16.
// All buffers < 4GB: divergent addresses are formed as uniform base + zext u32
// offset so the backend selects SADDR (GVS) addressing, killing 64-bit VALU
// address chains that compete with the WMMA issue stream.
// ---------------------------------------------------------------------------

#define N_VOX  200000
#define A_CH   128
#define BC_CH  256
#define K_OFF  27
#define BR_N   8
#define EPSV   1e-4f
#define TILES  (N_VOX / 16)      // 12500, exact
#define LDSROW 132               // row stride (132 % 64 = 4 -> conflict-free column reads)
#define ROWB   (A_CH * 4)        // 512 bytes per H row
#define OUTB   (BC_CH * 4)       // 1024 bytes per out row

typedef __attribute__((ext_vector_type(2))) float v2f;
typedef __attribute__((ext_vector_type(8))) float v8f;

// ws float layout:
//   [0]    xsum[128]     [128]  xssq[128]    [256] xscale[128]  [384] xshift[128]
//   [512]  h1sum[128]    [640]  h1ssq[128]   [768] h1scale[128] [896] h1shift[128]
//   [1024] h2sum[128]    [1152] h2ssq[128]   [1280]h2scale[128] [1408]h2shift[128]
//   [1536] bosum[256]
//   [2048] H1[N*128]     [2048+N*128] H2[N*128]
#define WS_H1_OFF 2048
#define WS_H2_OFF (2048 + (size_t)N_VOX * 128)

__device__ __forceinline__ v8f wmma4(v2f a, v2f b, v8f c) {
  // D(16x16,f32) += A(16x4,f32) * B(4x16,f32)   -- v_wmma_f32_16x16x4_f32
  return __builtin_amdgcn_wmma_f32_16x16x4_f32(
      /*neg_a=*/false, a, /*neg_b=*/false, b,
      /*c_mod=*/(short)0, c, /*reuse_a=*/false, /*reuse_b=*/false);
}

// --------------------------- async-to-LDS helpers --------------------------
#if defined(__gfx1250__) && __has_builtin(__builtin_amdgcn_global_load_async_to_lds_b128) && \
    __has_builtin(__builtin_amdgcn_global_load_async_to_lds_b32)
#define HAVE_ASYNC_LDS 1
typedef int v4i __attribute__((vector_size(16)));              // matches builtin pointee
typedef __attribute__((address_space(1))) v4i gv4i;            // global int4
typedef __attribute__((address_space(3))) v4i lv4i;            // LDS int4
typedef __attribute__((address_space(1))) int gi32;            // global int
typedef __attribute__((address_space(3))) int li32;            // LDS int

// global flat address == AS1 address numerically; LDS generic low 32 bits == LDS offset
__device__ __forceinline__ void async_g2l_b128(const void* g, void* l) {
  __builtin_amdgcn_global_load_async_to_lds_b128(
      (gv4i*)(uintptr_t)g, (lv4i*)(uint32_t)(uintptr_t)l, 0, 0);
}
__device__ __forceinline__ void async_g2l_b32(const void* g, void* l) {
  __builtin_amdgcn_global_load_async_to_lds_b32(
      (gi32*)(uintptr_t)g, (li32*)(uint32_t)(uintptr_t)l, 0, 0);
}
#else
#define HAVE_ASYNC_LDS 0
#endif

__device__ __forceinline__ void wait_async0() {
#if defined(__gfx1250__)
#if __has_builtin(__builtin_amdgcn_s_wait_asynccnt)
  __builtin_amdgcn_s_wait_asynccnt(0);
#else
  asm volatile("s_wait_asynccnt 0" ::: "memory");
#endif
#endif
}

// ---------------------------------------------------------------------------
__global__ void kzero(float* ws) {
  int i = blockIdx.x * 256 + threadIdx.x;
  if (i < 2048) ws[i] = 0.f;
}

// per-channel sum / sumsq of x; register-accumulate, one atomic per thread
__global__ void kcolstats(const float* __restrict__ x, float* __restrict__ sum,
                          float* __restrict__ ssq) {
  int c = threadIdx.x;  // 128 threads
  const char* xb = (const char*)x;
  float s = 0.f, q = 0.f;
  for (int r = blockIdx.x; r < N_VOX; r += gridDim.x) {
    float v = *(const float*)(xb + (uint32_t)r * ROWB + (uint32_t)c * 4);
    s += v;
    q += v * v;
  }
  atomicAdd(&sum[c], s);
  atomicAdd(&ssq[c], q);
}

// BN finalize: scale = g*rsqrt(var+eps), shift = b - mean*scale  (128 threads)
__global__ void kfinalize(const float* __restrict__ sum, const float* __restrict__ ssq,
                          const float* __restrict__ g, const float* __restrict__ b,
                          float* __restrict__ scale, float* __restrict__ shift) {
  int c = threadIdx.x;
  const float invN = 1.0f / (float)N_VOX;
  float m = sum[c] * invN;
  float v = ssq[c] * invN - m * m;
  float s = g[c] * rsqrtf(v + EPSV);
  scale[c] = s;
  shift[c] = b[c] - m * s;
}

__global__ void kbias(const float* __restrict__ bo, float* __restrict__ bosum) {
  int c = threadIdx.x;  // 256 threads
  float s = 0.f;
#pragma unroll
  for (int br = 0; br < BR_N; br++) s += bo[br * BC_CH + c];
  bosum[c] = s;
}

// ---------------------------------------------------------------------------
// k2: per 16-row tile: out = x @ Wdir (256 cols), H1 = relu(bn(x)) @ Wl + bl.
// 24 waves / 768 threads: waves 0..15 -> one out tile each; waves 16..23 -> one
// H1 branch each. One hoisted B-fragment set (64 VGPRs) per wave: no spills.
__global__ void __launch_bounds__(768) k2_fused_in(
    const float* __restrict__ x, const float* __restrict__ Wdir,
    const float* __restrict__ Wl, const float* __restrict__ bl,
    const float* __restrict__ xscale, const float* __restrict__ xshift,
    float* __restrict__ H1, float* __restrict__ out,
    float* __restrict__ h1sum, float* __restrict__ h1ssq) {
  __shared__ float xs[16 * LDSROW];
  __shared__ float bs[16 * LDSROW];
  const int tid = threadIdx.x;
  const int lane = tid & 31, wave = tid >> 5;  // 24 waves
  const int row = lane & 15, hi = lane >> 4, col = lane & 15;
  const bool isH1 = (wave >= 16);
  const int br = wave - 16;  // valid when isH1
  const int t = wave;        // out tile when wave < 16
  char* outb = (char*)out;
  char* h1b = (char*)H1;
  float ssum = 0.f, sssq = 0.f;

  for (int tile = blockIdx.x; tile < TILES; tile += gridDim.x) {
    __syncthreads();  // protect LDS reuse across grid-stride iterations
    const int row0 = tile * 16;
#if HAVE_ASYNC_LDS
    if (tid < 512) {  // 512 x b128 = 16x128 floats straight into LDS
      int r = tid >> 5, c4 = (tid & 31) * 4;
      async_g2l_b128(x + (size_t)(row0 + r) * A_CH + c4, &xs[r * LDSROW + c4]);
    }
    wait_async0();
    __syncthreads();
    for (int i = tid; i < 16 * A_CH; i += 768) {
      int r = i >> 7, c = i & 127;
      bs[r * LDSROW + c] = fmaxf(fmaf(xs[r * LDSROW + c], xscale[c], xshift[c]), 0.f);
    }
#else
    for (int i = tid; i < 16 * A_CH; i += 768) {
      int r = i >> 7, c = i & 127;
      float v = x[(size_t)(row0 + r) * A_CH + c];
      xs[r * LDSROW + c] = v;
      bs[r * LDSROW + c] = fmaxf(fmaf(v, xscale[c], xshift[c]), 0.f);
    }
#endif
    __syncthreads();

    if (!isH1) {
      // direct path: out tile t
      v8f d = {};
      for (int k0 = 0; k0 < A_CH; k0 += 4) {
        v2f a;
        a.x = xs[row * LDSROW + k0 + 2 * hi];
        a.y = xs[row * LDSROW + k0 + 2 * hi + 1];
        v2f b;
        b.x = Wdir[(size_t)(k0 + 2 * hi) * BC_CH + t * 16 + col];
        b.y = Wdir[(size_t)(k0 + 2 * hi + 1) * BC_CH + t * 16 + col];
        d = wmma4(a, b, d);
      }
      const uint32_t ob = (uint32_t)(row0 + 8 * hi) * OUTB + (uint32_t)(t * 16 + col) * 4;
#pragma unroll
      for (int r = 0; r < 8; r++)
        *(float*)(outb + ob + (uint32_t)r * OUTB) = d[r];
    } else {
      // branch linear: H1 tile (cols br*16 .. br*16+15)
      v8f d = {};
      for (int k0 = 0; k0 < A_CH; k0 += 4) {
        v2f a;
        a.x = bs[row * LDSROW + k0 + 2 * hi];
        a.y = bs[row * LDSROW + k0 + 2 * hi + 1];
        v2f b;
        b.x = Wl[(size_t)(br * A_CH + k0 + 2 * hi) * 16 + col];
        b.y = Wl[(size_t)(br * A_CH + k0 + 2 * hi + 1) * 16 + col];
        d = wmma4(a, b, d);
      }
      const float bias = bl[br * 16 + col];
      const uint32_t ob = (uint32_t)(row0 + 8 * hi) * ROWB + (uint32_t)(br * 16 + col) * 4;
#pragma unroll
      for (int r = 0; r < 8; r++) {
        float v = d[r] + bias;
        *(float*)(h1b + ob + (uint32_t)r * ROWB) = v;
        ssum += v;
        sssq += v * v;
      }
    }
  }
  if (isH1) {
    atomicAdd(&h1sum[br * 16 + col], ssum);
    atomicAdd(&h1ssq[br * 16 + col], sssq);
  }
}

// ---------------------------------------------------------------------------
// submanifold conv fragment for one branch: D(16x16) = sum_k gather(bnrelu(H)) * W[k]
// nbrs = LDS-staged 16x27 neighbor indices; sentinel idx==N -> zero contribution.
// Gathers batched per offset; next offset's row prefetched while WMMAs issue.
// All addresses: uniform base + u32 offset (SADDR form) + immediate deltas.
__device__ __forceinline__ v8f subconv_frag(const float* __restrict__ H,
                                            const float* __restrict__ W,
                                            const int* nbrs,
                                            const float* __restrict__ scale,
                                            const float* __restrict__ shift,
                                            int br, int row, int hi, int col) {
  v2f sc[4], sh[4];
#pragma unroll
  for (int c = 0; c < 4; c++) {
    int ch = br * 16 + c * 4 + 2 * hi;
    sc[c].x = scale[ch];
    sc[c].y = scale[ch + 1];
    sh[c].x = shift[ch];
    sh[c].y = shift[ch + 1];
  }
  const char* Hb = (const char*)H;
  const char* Wb = (const char*)W;
  const uint32_t chb = (uint32_t)(br * 16) * 4;  // branch channel byte offset
  v8f d = {};
  int idx = nbrs[row * K_OFF];
  for (int k = 0; k < K_OFF; k++) {
    const bool valid = (idx < N_VOX);
    // gather row base: u32 byte offset from uniform H; per-chunk delta is an imm
    const uint32_t roff = (uint32_t)(valid ? idx : 0) * ROWB + chb + (uint32_t)hi * 8;
    v2f g[4];
#pragma unroll
    for (int c = 0; c < 4; c++) g[c] = *(const v2f*)(Hb + roff + c * 16);
    if (k + 1 < K_OFF) {  // prefetch next offset's gather row (uniform branch)
      idx = nbrs[row * K_OFF + k + 1];
      uint32_t poff = (uint32_t)((idx < N_VOX) ? idx : 0) * ROWB + chb;
      __builtin_prefetch((const void*)(Hb + poff), 0, 3);
    }
    // weight block: W[(br*27+k)*256 + (c*4+2hi)*16 + col] -> imm deltas per chunk
    const uint32_t woff =
        (uint32_t)(br * K_OFF + k) * 1024 + (uint32_t)hi * 128 + (uint32_t)col * 4;
#pragma unroll
    for (int c = 0; c < 4; c++) {
      v2f a;
      a.x = valid ? fmaxf(fmaf(g[c].x, sc[c].x, sh[c].x), 0.f) : 0.f;
      a.y = valid ? fmaxf(fmaf(g[c].y, sc[c].y, sh[c].y), 0.f) : 0.f;
      v2f b;
      b.x = *(const float*)(Wb + woff + c * 256);
      b.y = *(const float*)(Wb + woff + c * 256 + 64);
      d = wmma4(a, b, d);
    }
  }
  return d;
}

// nbr tile staging (16 x 27 ints), async when available
__device__ __forceinline__ void stage_nbr(const int* __restrict__ nbr, int* nb,
                                          int row0, int tid) {
#if HAVE_ASYNC_LDS
  for (int i = tid; i < 16 * K_OFF; i += 256)
    async_g2l_b32(nbr + (size_t)row0 * K_OFF + i, &nb[i]);
  wait_async0();
#else
  for (int i = tid; i < 16 * K_OFF; i += 256) nb[i] = nbr[(size_t)row0 * K_OFF + i];
#endif
}

// k3: H2 = subconv(bnrelu(H1), W1) + stats
__global__ void __launch_bounds__(256) k3_subconv1(
    const float* __restrict__ H1, const float* __restrict__ W1,
    const int* __restrict__ nbr, const float* __restrict__ sc,
    const float* __restrict__ sh, float* __restrict__ H2,
    float* __restrict__ h2sum, float* __restrict__ h2ssq) {
  __shared__ int nb[16 * K_OFF];
  const int tid = threadIdx.x;
  const int lane = tid & 31, wave = tid >> 5;
  const int row = lane & 15, hi = lane >> 4, col = lane & 15;
  const int br = wave;
  char* h2b = (char*)H2;
  float ssum = 0.f, sssq = 0.f;

  for (int tile = blockIdx.x; tile < TILES; tile += gridDim.x) {
    __syncthreads();
    const int row0 = tile * 16;
    stage_nbr(nbr, nb, row0, tid);
    __syncthreads();

    v8f d = subconv_frag(H1, W1, nb, sc, sh, br, row, hi, col);
    const uint32_t ob = (uint32_t)(row0 + 8 * hi) * ROWB + (uint32_t)(br * 16 + col) * 4;
#pragma unroll
    for (int r = 0; r < 8; r++) {
      float v = d[r];
      *(float*)(h2b + ob + (uint32_t)r * ROWB) = v;
      ssum += v;
      sssq += v * v;
    }
  }
  atomicAdd(&h2sum[br * 16 + col], ssum);
  atomicAdd(&h2ssq[br * 16 + col], sssq);
}

// k4: H3 = subconv(bnrelu(H2), W2) -> LDS; then out += H3 @ Wo_stack + sum(bo)
__global__ void __launch_bounds__(256) k4_subconv2_out(
    const float* __restrict__ H2, const float* __restrict__ W2,
    const int* __restrict__ nbr, const float* __restrict__ Wo,
    const float* __restrict__ sc, const float* __restrict__ sh,
    const float* __restrict__ bosum, float* __restrict__ out) {
  __shared__ int nb[16 * K_OFF];
  __shared__ float h3[16 * LDSROW];
  const int tid = threadIdx.x;
  const int lane = tid & 31, wave = tid >> 5;
  const int row = lane & 15, hi = lane >> 4, col = lane & 15;
  const int br = wave;
  char* outb = (char*)out;

  for (int tile = blockIdx.x; tile < TILES; tile += gridDim.x) {
    __syncthreads();
    const int row0 = tile * 16;
    stage_nbr(nbr, nb, row0, tid);
    __syncthreads();

    v8f d = subconv_frag(H2, W2, nb, sc, sh, br, row, hi, col);
#pragma unroll
    for (int r = 0; r < 8; r++) h3[(r + 8 * hi) * LDSROW + br * 16 + col] = d[r];
    __syncthreads();

    // out tiles t and t+8: out += h3 @ Wo_stack + bosum
    for (int t = wave; t < 16; t += 8) {
      v8f acc = {};
#pragma unroll 4  // keep Wo fragments streaming from L1/L2, not hoisted into spills
      for (int k0 = 0; k0 < A_CH; k0 += 4) {
        v2f a;
        a.x = h3[row * LDSROW + k0 + 2 * hi];
        a.y = h3[row * LDSROW + k0 + 2 * hi + 1];
        v2f b;
        b.x = Wo[(size_t)(k0 + 2 * hi) * BC_CH + t * 16 + col];
        b.y = Wo[(size_t)(k0 + 2 * hi + 1) * BC_CH + t * 16 + col];
        acc = wmma4(a, b, acc);
      }
      const float bb = bosum[t * 16 + col];
      const uint32_t ob = (uint32_t)(row0 + 8 * hi) * OUTB + (uint32_t)(t * 16 + col) * 4;
#pragma unroll
      for (int r = 0; r < 8; r++) {
        float* p = (float*)(outb + ob + (uint32_t)r * OUTB);
        *p = *p + acc[r] + bb;
      }
    }
  }
}

// ---------------------------------------------------------------------------
extern "C" void kernel_launch(void* const* d_in, const int* in_sizes, int n_in,
                              void* d_out, int out_size, void* d_ws, size_t ws_size,
                              hipStream_t stream) {
  (void)in_sizes; (void)n_in; (void)out_size; (void)ws_size;
  const float* x    = (const float*)d_in[0];
  const int*   nbr  = (const int*)d_in[1];
  const float* Wdir = (const float*)d_in[2];
  const float* gbn  = (const float*)d_in[3];
  const float* bbn  = (const float*)d_in[4];
  const float* Wl   = (const float*)d_in[5];
  const float* bl   = (const float*)d_in[6];
  const float* g1   = (const float*)d_in[7];
  const float* b1   = (const float*)d_in[8];
  const float* W1   = (const float*)d_in[9];
  const float* g2   = (const float*)d_in[10];
  const float* b2   = (const float*)d_in[11];
  const float* W2   = (const float*)d_in[12];
  const float* Wo   = (const float*)d_in[13];
  const float* bo   = (const float*)d_in[14];
  float* out = (float*)d_out;
  float* ws = (float*)d_ws;
  float* H1 = ws + WS_H1_OFF;
  float* H2 = ws + WS_H2_OFF;

  kzero<<<8, 256, 0, stream>>>(ws);
  kcolstats<<<512, 128, 0, stream>>>(x, ws + 0, ws + 128);
  kfinalize<<<1, 128, 0, stream>>>(ws + 0, ws + 128, gbn, bbn, ws + 256, ws + 384);
  k2_fused_in<<<1024, 768, 0, stream>>>(x, Wdir, Wl, bl, ws + 256, ws + 384,
                                        H1, out, ws + 512, ws + 640);
  kfinalize<<<1, 128, 0, stream>>>(ws + 512, ws + 640, g1, b1, ws + 768, ws + 896);
  k3_subconv1<<<2048, 256, 0, stream>>>(H1, W1, nbr, ws + 768, ws + 896,
                                        H2, ws + 1024, ws + 1152);
  kfinalize<<<1, 128, 0, stream>>>(ws + 1024, ws + 1152, g2, b2, ws + 1280, ws + 1408);
  kbias<<<1, 256, 0, stream>>>(bo, ws + 1536);
  k4_subconv2_out<<<2048, 256, 0, stream>>>(H2, W2, nbr, Wo, ws + 1280, ws + 1408,
                                            ws + 1536, out);
}